// VSRF_69784628626149
// MI455X (gfx1250) — compile-verified
//
#include <hip/hip_runtime.h>
#include <hip/hip_bf16.h>
#include <cstdint>

#define BB 8
#define SS 8
#define PP 5
#define KTOP 3
#define CC 2048
#define DD 512
#define D2 1024
#define VV 10000
#define BS (BB*SS)

typedef float v2f __attribute__((ext_vector_type(2)));
typedef float v8f __attribute__((ext_vector_type(8)));

// ---------------------------------------------------------------------------
// Kernel A: per-(b,s) top-5 over V=10000 word scores.
// Stages the 40KB row into LDS with CDNA5 GLOBAL_LOAD_ASYNC_TO_LDS_B32
// (ASYNCcnt path), then each thread keeps a sorted local top-5 and thread 0
// merges 256*5 candidates (tie-break: lower word index, matching lax.top_k).
// ---------------------------------------------------------------------------
__global__ void topp_kernel(const float* __restrict__ predicts,
                            float* __restrict__ ovals, int* __restrict__ oidx) {
  __shared__ float row[VV];
  __shared__ float cval[256 * PP];
  __shared__ int   cidx[256 * PP];
  const int bs = blockIdx.x, tid = threadIdx.x;
  const float* src = predicts + (size_t)bs * VV;

  for (int i = tid; i < VV; i += 256) {
    uint32_t lds = (uint32_t)(uintptr_t)(&row[i]);
    uint64_t ga  = (uint64_t)(uintptr_t)(src + i);
    asm volatile("global_load_async_to_lds_b32 %0, %1, off"
                 :: "v"(lds), "v"(ga) : "memory");
  }
  asm volatile("s_wait_asynccnt 0" ::: "memory");
  __syncthreads();

  const float INF = __builtin_inff();
  float tv[PP]; int ti[PP];
  for (int k = 0; k < PP; ++k) { tv[k] = -INF; ti[k] = VV; }
  for (int i = tid; i < VV; i += 256) {
    float v = row[i];
    if (v > tv[PP - 1]) {            // strict > : equal values keep lower index
      tv[PP - 1] = v; ti[PP - 1] = i;
      for (int k = PP - 1; k > 0; --k) {
        if (tv[k] > tv[k - 1]) {
          float fv = tv[k]; tv[k] = tv[k - 1]; tv[k - 1] = fv;
          int   iv = ti[k]; ti[k] = ti[k - 1]; ti[k - 1] = iv;
        } else break;
      }
    }
  }
  for (int k = 0; k < PP; ++k) { cval[tid * PP + k] = tv[k]; cidx[tid * PP + k] = ti[k]; }
  __syncthreads();

  if (tid == 0) {
    for (int p = 0; p < PP; ++p) {
      float bv = -INF; int bi = VV; int sel = 0;
      for (int t = 0; t < 256 * PP; ++t) {
        float v = cval[t]; int id = cidx[t];
        if (v > bv || (v == bv && id < bi)) { bv = v; bi = id; sel = t; }
      }
      cval[sel] = -INF;
      ovals[bs * PP + p] = bv;
      oidx [bs * PP + p] = bi;
    }
  }
}

// ---------------------------------------------------------------------------
// Kernel B: per-cluster row sums sumW[c] = W[c,:] . ones  (true GEMV) via
// V_WMMA_F32_16X16X4_F32. One wave handles 16 clusters; A = 16x4 tile of W
// (lanes 0-15: K=k,k+1 ; lanes 16-31: K=k+2,k+3), B = all-ones 4x16.
// After 256 WMMAs, D[m, n] = sumW[c0+m] for every n; lanes 0 and 16 (N=0)
// hold M=0..7 / M=8..15 in accumulator VGPRs 0..7.
// ---------------------------------------------------------------------------
__global__ void sumw_kernel(const float* __restrict__ W, float* __restrict__ sumW) {
  const int lane = threadIdx.x;
  const int c0   = blockIdx.x * 16;
  const int m    = lane & 15;
  const int kof  = (lane >> 4) << 1;          // 0 or 2
  v8f acc = {0.f,0.f,0.f,0.f,0.f,0.f,0.f,0.f};
  v2f ones; ones[0] = 1.0f; ones[1] = 1.0f;
  const float* rowp = W + (size_t)(c0 + m) * D2 + kof;
  for (int k = 0; k < D2; k += 4) {
    v2f a; a[0] = rowp[k]; a[1] = rowp[k + 1];
    acc = __builtin_amdgcn_wmma_f32_16x16x4_f32(
        false, a, false, ones, (short)0, acc, false, false);
  }
  if (lane == 0)  for (int r = 0; r < 8; ++r) sumW[c0 + r]     = acc[r];
  if (lane == 16) for (int r = 0; r < 8; ++r) sumW[c0 + 8 + r] = acc[r];
}

// ---------------------------------------------------------------------------
// Kernel C: sim[bs,c] = inter / (512 + sumW[c] - inter),
//           inter = sum_d min(H[bs,d], W[c,d]) over 2D=1024 dims.
// All 64 scaled H vectors (512 floats each) live in 128KB LDS (CDNA5 320KB
// WGP capacity); complement half is computed on the fly (H[d+512]=1-H[d]).
// Each block owns a 32-cluster tile -> W is streamed from HBM exactly once.
// ---------------------------------------------------------------------------
__global__ void sim_kernel(const float* __restrict__ hv,
                           const float* __restrict__ W,
                           const float* __restrict__ fmin,
                           const float* __restrict__ fmax,
                           const float* __restrict__ sumW,
                           float* __restrict__ sim) {
  extern __shared__ float Hs[];               // [BS][DD] = 128 KB
  const int tid = threadIdx.x;
  for (int i = tid; i < BS * DD; i += 256) {
    int bs = i >> 9, dd = i & (DD - 1);
    float mn = fmin[dd], mx = fmax[dd];
    Hs[i] = (hv[bs * DD + dd] - mn) / (mx - mn);
  }
  __syncthreads();

  const int wave = tid >> 5, lane = tid & 31;
  for (int ci = 0; ci < 4; ++ci) {
    const int c = blockIdx.x * 32 + wave * 4 + ci;
    const float sw = sumW[c];
    const float* wr = W + (size_t)c * D2;
    for (int g = 0; g < 8; ++g) {             // 8 bs values at a time
      float acc[8] = {0,0,0,0,0,0,0,0};
      for (int j = 0; j < DD / 32; ++j) {
        int dd = lane + 32 * j;               // coalesced, LDS conflict-free
        float w1 = wr[dd], w2 = wr[DD + dd];
        #pragma unroll
        for (int r = 0; r < 8; ++r) {
          float s = Hs[(g * 8 + r) * DD + dd];
          acc[r] += fminf(s, w1) + fminf(1.0f - s, w2);
        }
      }
      #pragma unroll
      for (int r = 0; r < 8; ++r) {
        float v = acc[r];
        for (int off = 16; off > 0; off >>= 1) v += __shfl_xor(v, off, 32);
        if (lane == 0) {
          float uni = (float)DD + sw - v;     // sumH == 512 exactly
          sim[(size_t)(g * 8 + r) * CC + c] = v / uni;
        }
      }
    }
  }
}

// ---------------------------------------------------------------------------
// Kernel D: per (b,s): for each of the 5 candidate words, masked argmax of
// sim over clusters (mask = cli[c,word]>0, fallback to unmasked if empty),
// then winner = argmax_p best_sim*val. Emits decision word, class_sim, label.
// ---------------------------------------------------------------------------
__global__ void decide_kernel(const float* __restrict__ sim,
                              const int*   __restrict__ cli,
                              const int*   __restrict__ vcc,
                              const float* __restrict__ vals,
                              const int*   __restrict__ widx,
                              float* __restrict__ out,
                              float* __restrict__ wlabel) {
  __shared__ float mval[256]; __shared__ int mcid[256];
  __shared__ float aval[256]; __shared__ int acid[256];
  __shared__ int   anyf[256];
  const int bs = blockIdx.x, tid = threadIdx.x;
  const float INF = __builtin_inff();
  float bestF = -INF; int winWord = 0; int winCls = 0;

  for (int p = 0; p < PP; ++p) {
    int word = widx[bs * PP + p];
    float bm = -INF; int bmc = CC; float ba = -INF; int bac = CC; int any = 0;
    for (int c = tid; c < CC; c += 256) {
      float sv = sim[(size_t)bs * CC + c];
      if (cli[(size_t)c * VV + word] > 0) {
        any = 1;
        if (sv > bm) { bm = sv; bmc = c; }    // strict > : lowest c wins ties
      }
      if (sv > ba) { ba = sv; bac = c; }
    }
    mval[tid] = bm; mcid[tid] = bmc; aval[tid] = ba; acid[tid] = bac; anyf[tid] = any;
    __syncthreads();
    if (tid == 0) {
      int anyAll = 0; float gm = -INF; int gmc = CC; float ga = -INF; int gac = CC;
      for (int t = 0; t < 256; ++t) {
        anyAll |= anyf[t];
        if (mval[t] > gm || (mval[t] == gm && mcid[t] < gmc)) { gm = mval[t]; gmc = mcid[t]; }
        if (aval[t] > ga || (aval[t] == ga && acid[t] < gac)) { ga = aval[t]; gac = acid[t]; }
      }
      float bsim = anyAll ? gm  : ga;
      int   bc   = anyAll ? gmc : gac;
      float f = bsim * vals[bs * PP + p];
      if (f > bestF) { bestF = f; winWord = word; winCls = vcc[bc]; }  // first max wins
    }
    __syncthreads();
  }
  if (tid == 0) {
    out[bs] = (float)winWord;                 // decision_words
    out[BS + BB * KTOP + bs] = bestF;         // class_sim
    wlabel[bs] = (float)winCls;
  }
}

// ---------------------------------------------------------------------------
// Kernel E: per batch b, top-3 of class_sim over S=8, emit their labels.
// ---------------------------------------------------------------------------
__global__ void topcls_kernel(float* __restrict__ out, const float* __restrict__ wlabel) {
  int b = threadIdx.x;
  if (b >= BB) return;
  const float* cs = out + BS + BB * KTOP + b * SS;
  const float* lb = wlabel + b * SS;
  bool used[SS] = {};
  for (int t = 0; t < KTOP; ++t) {
    float best = -__builtin_inff(); int bi = 0;
    for (int s = 0; s < SS; ++s)
      if (!used[s] && cs[s] > best) { best = cs[s]; bi = s; }  // ties: lower s
    used[bi] = true;
    out[BS + b * KTOP + t] = lb[bi];          // decision_classes_topk
  }
}

// ---------------------------------------------------------------------------
extern "C" void kernel_launch(void* const* d_in, const int* in_sizes, int n_in,
                              void* d_out, int out_size, void* d_ws, size_t ws_size,
                              hipStream_t stream) {
  (void)in_sizes; (void)n_in; (void)out_size; (void)ws_size;
  const float* predicts = (const float*)d_in[0];
  const float* hv       = (const float*)d_in[1];
  const float* W        = (const float*)d_in[2];
  const int*   cli      = (const int*)  d_in[3];
  const int*   vcc      = (const int*)  d_in[4];
  const float* fmin     = (const float*)d_in[5];
  const float* fmax     = (const float*)d_in[6];
  float* out = (float*)d_out;

  // workspace carve-up (~536 KB)
  float* sim    = (float*)d_ws;               // [64][2048]
  float* sumW   = sim + (size_t)BS * CC;      // [2048]
  float* vals   = sumW + CC;                  // [64][5]
  int*   widx   = (int*)(vals + BS * PP);     // [64][5]
  float* wlabel = (float*)(widx + BS * PP);   // [64]

  topp_kernel<<<BS, 256, 0, stream>>>(predicts, vals, widx);
  sumw_kernel<<<CC / 16, 32, 0, stream>>>(W, sumW);

  size_t shmem = (size_t)BS * DD * sizeof(float);   // 128 KB dynamic LDS
  hipFuncSetAttribute(reinterpret_cast<const void*>(sim_kernel),
                      hipFuncAttributeMaxDynamicSharedMemorySize, (int)shmem);
  sim_kernel<<<CC / 32, 256, shmem, stream>>>(hv, W, fmin, fmax, sumW, sim);

  decide_kernel<<<BS, 256, 0, stream>>>(sim, cli, vcc, vals, widx, out, wlabel);
  topcls_kernel<<<1, 32, 0, stream>>>(out, wlabel);
}